// MagnetLoss_75179107549355
// MI455X (gfx1250) — compile-verified
//
#include <hip/hip_runtime.h>
#include <hip/hip_bf16.h>

typedef __attribute__((ext_vector_type(2))) float v2f;
typedef __attribute__((ext_vector_type(4))) float v4f;
typedef __attribute__((ext_vector_type(8))) float v8f;

// Problem constants
#define NTOT   16384      // M * D_PER samples
#define FEATN  1024
#define NCLUST 512
#define DPER   32
#define TK     32         // K tile staged in LDS
#define LDP    36         // padded K stride in LDS (16B-aligned, bank-friendly)

// Workspace layout (float offsets)
#define OFF_MEANS  0ul                      // 512*1024
#define OFF_C2     524288ul                 // 512
#define OFF_R2     524800ul                 // 16384
#define OFF_COSTS  541184ul                 // 16384*512
#define OFF_INTRA  8929792ul                // 16384
#define OFF_PRED   8946176ul                // 16384 (ints)
#define OFF_PINTRA 8962560ul                // 2048
#define OFF_PACC   8964608ul                // 2048
#define OFF_PLOSS  8966656ul                // 2048
#define OFF_SCAL   8968704ul                // 4

// ---------------------------------------------------------------- means + c2
__global__ __launch_bounds__(256)
void means_kernel(const float* __restrict__ inp, float* __restrict__ means,
                  float* __restrict__ c2) {
    __shared__ float red[256];
    const int c   = blockIdx.x;
    const int tid = threadIdx.x;
    const float* base = inp + (size_t)c * DPER * FEATN + tid * 4;
    float s0 = 0.f, s1 = 0.f, s2 = 0.f, s3 = 0.f;
    #pragma unroll 8
    for (int r = 0; r < DPER; ++r) {
        v4f v = *(const v4f*)(base + (size_t)r * FEATN);
        s0 += v.x; s1 += v.y; s2 += v.z; s3 += v.w;
    }
    const float inv = 1.0f / (float)DPER;
    s0 *= inv; s1 *= inv; s2 *= inv; s3 *= inv;
    v4f mv = {s0, s1, s2, s3};
    *(v4f*)(means + (size_t)c * FEATN + tid * 4) = mv;
    red[tid] = s0*s0 + s1*s1 + s2*s2 + s3*s3;
    __syncthreads();
    for (int off = 128; off; off >>= 1) {
        if (tid < off) red[tid] += red[tid + off];
        __syncthreads();
    }
    if (tid == 0) c2[c] = red[0];
}

// ---------------------------------------------------------------- r2 per row
__global__ __launch_bounds__(256)
void r2_kernel(const float* __restrict__ inp, float* __restrict__ r2p) {
    const int wave = threadIdx.x >> 5, lane = threadIdx.x & 31;
    const int row  = blockIdx.x * 8 + wave;
    const v4f* p = (const v4f*)(inp + (size_t)row * FEATN);
    float s = 0.f;
    #pragma unroll
    for (int t = 0; t < 8; ++t) {
        v4f v = p[lane + t * 32];
        s += v.x*v.x + v.y*v.y + v.z*v.z + v.w*v.w;
    }
    for (int off = 16; off; off >>= 1) s += __shfl_xor(s, off, 32);
    if (lane == 0) r2p[row] = s;
}

// ------------------------------------------- costs = r2 + c2 - 2*inp@meansT
// Block tile: 128 rows x 64 cols, 8 waves (4 row-groups x 2 col-groups).
// Each wave owns 4 f32 16x16 accumulators (2 row-tiles x 2 col-tiles):
// 4 WMMAs per 4 ds_load_b64 fragment fetches.
// Both LDS tiles are K-contiguous ([row/col][k], stride LDP) so every WMMA
// fragment (K pair {2h, 2h+1}) is one aligned 8-byte LDS load.
__global__ __launch_bounds__(256)
void costs_kernel(const float* __restrict__ inp, const float* __restrict__ means,
                  const float* __restrict__ r2p, const float* __restrict__ c2,
                  float* __restrict__ costs) {
    __shared__ float As[128][LDP];
    __shared__ float Bs[64][LDP];
    const int tid  = threadIdx.x;
    const int wave = tid >> 5;
    const int lane = tid & 31;
    const int half = lane >> 4;    // K-pair selector in WMMA frags
    const int sub  = lane & 15;    // M (for A/D) or N (for B/D) within tile
    const int rowBlock = blockIdx.x * 128;
    const int colBlock = blockIdx.y * 64;
    const int waveRow = (wave & 3) * 32;   // 0,32,64,96
    const int waveCol = (wave >> 2) * 32;  // 0,32

    // A fill: thread -> (row 0..127, 16 consecutive k)
    const int ar = tid >> 1;
    const int ak = (tid & 1) * 16;
    // B fill: thread -> (col 0..63, 8 consecutive k)
    const int br = tid >> 2;
    const int bk = (tid & 3) * 8;
    const float* aG = inp   + (size_t)(rowBlock + ar) * FEATN + ak;
    const float* bG = means + (size_t)(colBlock + br) * FEATN + bk;

    v8f acc00 = {}, acc01 = {}, acc10 = {}, acc11 = {};

    for (int kb = 0; kb < FEATN; kb += TK) {
        v4f a0 = *(const v4f*)(aG + kb);
        v4f a1 = *(const v4f*)(aG + kb + 4);
        v4f a2 = *(const v4f*)(aG + kb + 8);
        v4f a3 = *(const v4f*)(aG + kb + 12);
        v4f b0 = *(const v4f*)(bG + kb);
        v4f b1 = *(const v4f*)(bG + kb + 4);
        __syncthreads();
        *(v4f*)&As[ar][ak]      = a0;
        *(v4f*)&As[ar][ak + 4]  = a1;
        *(v4f*)&As[ar][ak + 8]  = a2;
        *(v4f*)&As[ar][ak + 12] = a3;
        *(v4f*)&Bs[br][bk]      = b0;
        *(v4f*)&Bs[br][bk + 4]  = b1;
        __syncthreads();
        if (kb + TK < FEATN) {   // hint next K tile toward caches (uniform branch)
            __builtin_prefetch(aG + kb + TK, 0, 0);
            __builtin_prefetch(bG + kb + TK, 0, 0);
        }
        #pragma unroll
        for (int ks = 0; ks < TK; ks += 4) {
            const int kf = ks + 2 * half;
            // A frags (16x4): lane -> m = sub, K = {kf, kf+1}
            v2f aF0 = *(const v2f*)&As[waveRow + sub][kf];
            v2f aF1 = *(const v2f*)&As[waveRow + 16 + sub][kf];
            // B frags (4x16): lane -> n = sub, K = {kf, kf+1}
            v2f bF0 = *(const v2f*)&Bs[waveCol + sub][kf];
            v2f bF1 = *(const v2f*)&Bs[waveCol + 16 + sub][kf];
            acc00 = __builtin_amdgcn_wmma_f32_16x16x4_f32(
                false, aF0, false, bF0, (short)0, acc00, false, false);
            acc01 = __builtin_amdgcn_wmma_f32_16x16x4_f32(
                false, aF0, false, bF1, (short)0, acc01, false, false);
            acc10 = __builtin_amdgcn_wmma_f32_16x16x4_f32(
                false, aF1, false, bF0, (short)0, acc10, false, false);
            acc11 = __builtin_amdgcn_wmma_f32_16x16x4_f32(
                false, aF1, false, bF1, (short)0, acc11, false, false);
        }
    }

    // D layout: VGPR i -> M = i (lanes 0-15) / i+8 (lanes 16-31), N = sub
    #pragma unroll
    for (int i = 0; i < 8; ++i) {
        const int r0 = rowBlock + waveRow + i + 8 * half;
        const int r1 = r0 + 16;
        const int c0 = colBlock + waveCol + sub;
        const int c1 = c0 + 16;
        const float rr0 = r2p[r0];
        const float rr1 = r2p[r1];
        const float cc0 = c2[c0];
        const float cc1 = c2[c1];
        costs[(size_t)r0 * NCLUST + c0] = rr0 + cc0 - 2.0f * acc00[i];
        costs[(size_t)r0 * NCLUST + c1] = rr0 + cc1 - 2.0f * acc01[i];
        costs[(size_t)r1 * NCLUST + c0] = rr1 + cc0 - 2.0f * acc10[i];
        costs[(size_t)r1 * NCLUST + c1] = rr1 + cc1 - 2.0f * acc11[i];
    }
}

// ------------------------------------- per-row: intra, argmin(pred), partials
__global__ __launch_bounds__(256)
void rowpass_kernel(const float* __restrict__ costs, float* __restrict__ intra,
                    int* __restrict__ pred, float* __restrict__ pIntra,
                    float* __restrict__ pAcc) {
    __shared__ float sI[8];
    __shared__ float sA[8];
    const int wave = threadIdx.x >> 5, lane = threadIdx.x & 31;
    const int row = blockIdx.x * 8 + wave;
    const int cl  = row >> 5;            // row / D_PER
    const float* cr = costs + (size_t)row * NCLUST;
    float best = 3.402823466e38f;
    int bestj = 0;
    #pragma unroll
    for (int t = 0; t < 16; ++t) {
        const int j = lane + t * 32;     // increasing j per lane -> first min
        const float v = cr[j];
        if (v < best) { best = v; bestj = j; }
    }
    for (int off = 16; off; off >>= 1) {
        const float ov = __shfl_xor(best, off, 32);
        const int   oj = __shfl_xor(bestj, off, 32);
        if (ov < best || (ov == best && oj < bestj)) { best = ov; bestj = oj; }
    }
    if (lane == 0) {
        const float it = cr[cl];
        intra[row] = it;
        pred[row]  = bestj;
        sI[wave] = it;
        sA[wave] = (bestj == cl) ? 1.0f : 0.0f;
    }
    __syncthreads();
    if (threadIdx.x == 0) {
        float a = 0.f, b = 0.f;
        for (int i = 0; i < 8; ++i) { a += sI[i]; b += sA[i]; }
        pIntra[blockIdx.x] = a;
        pAcc[blockIdx.x]   = b;
    }
}

// ----------------------------------- variance -> var_norm, accuracy (1 block)
__global__ __launch_bounds__(256)
void variance_kernel(const float* __restrict__ pIntra, const float* __restrict__ pAcc,
                     float* __restrict__ scal, float* __restrict__ outAcc) {
    __shared__ float ra[256];
    __shared__ float rb[256];
    const int tid = threadIdx.x;
    float a = 0.f, b = 0.f;
    for (int i = tid; i < 2048; i += 256) { a += pIntra[i]; b += pAcc[i]; }
    ra[tid] = a; rb[tid] = b;
    __syncthreads();
    for (int off = 128; off; off >>= 1) {
        if (tid < off) { ra[tid] += ra[tid + off]; rb[tid] += rb[tid + off]; }
        __syncthreads();
    }
    if (tid == 0) {
        const float var = ra[0] / (float)(NTOT - 1);
        scal[0]   = -1.0f / (2.0f * var * var);
        outAcc[0] = rb[0] / (float)NTOT;
    }
}

// ------------------------------------------ per-row losses + pred-as-float
__global__ __launch_bounds__(256)
void loss_kernel(const float* __restrict__ costs, const int* __restrict__ target,
                 const float* __restrict__ intra, const int* __restrict__ pred,
                 const float* __restrict__ scal, float* __restrict__ outLoss,
                 float* __restrict__ outPred, float* __restrict__ pLoss) {
    __shared__ float sL[8];
    const int wave = threadIdx.x >> 5, lane = threadIdx.x & 31;
    const int row = blockIdx.x * 8 + wave;
    const float vn = scal[0];
    const int tgt = target[row];
    const float* cr = costs + (size_t)row * NCLUST;
    float den = 0.f;
    #pragma unroll
    for (int t = 0; t < 16; ++t) {
        const int j = lane + t * 32;
        const int cls = target[j << 5];          // class of cluster j
        if (cls != tgt) den += expf(vn * cr[j]);
    }
    for (int off = 16; off; off >>= 1) den += __shfl_xor(den, off, 32);
    if (lane == 0) {
        const float num = expf(vn * intra[row] - 1.0f);   // ALPHA = 1.0
        float l = -logf(num / (den + 1e-8f) + 1e-8f);
        l = l > 0.f ? l : 0.f;                            // relu
        outLoss[row] = l;
        outPred[row] = (float)pred[row];
        sL[wave] = l;
    }
    __syncthreads();
    if (threadIdx.x == 0) {
        float a = 0.f;
        for (int i = 0; i < 8; ++i) a += sL[i];
        pLoss[blockIdx.x] = a;
    }
}

// ---------------------------------------------------- total loss (1 block)
__global__ __launch_bounds__(256)
void finalize_kernel(const float* __restrict__ pLoss, float* __restrict__ outTotal) {
    __shared__ float r[256];
    const int tid = threadIdx.x;
    float a = 0.f;
    for (int i = tid; i < 2048; i += 256) a += pLoss[i];
    r[tid] = a;
    __syncthreads();
    for (int off = 128; off; off >>= 1) {
        if (tid < off) r[tid] += r[tid + off];
        __syncthreads();
    }
    if (tid == 0) outTotal[0] = r[0] / (float)NTOT;
}

extern "C" void kernel_launch(void* const* d_in, const int* in_sizes, int n_in,
                              void* d_out, int out_size, void* d_ws, size_t ws_size,
                              hipStream_t stream) {
    const float* inp    = (const float*)d_in[0];
    const int*   target = (const int*)d_in[1];
    float* out = (float*)d_out;
    float* w   = (float*)d_ws;

    float* means  = w + OFF_MEANS;
    float* c2     = w + OFF_C2;
    float* r2p    = w + OFF_R2;
    float* costs  = w + OFF_COSTS;
    float* intra  = w + OFF_INTRA;
    int*   pred   = (int*)(w + OFF_PRED);
    float* pIntra = w + OFF_PINTRA;
    float* pAcc   = w + OFF_PACC;
    float* pLoss  = w + OFF_PLOSS;
    float* scal   = w + OFF_SCAL;

    // Output layout: [0]=total_loss, [1..N]=losses, [N+1..2N]=pred, [2N+1]=acc
    float* outLoss  = out + 1;
    float* outPred  = out + 1 + NTOT;
    float* outAcc   = out + 1 + 2 * NTOT;

    means_kernel<<<NCLUST, 256, 0, stream>>>(inp, means, c2);
    r2_kernel<<<NTOT / 8, 256, 0, stream>>>(inp, r2p);
    dim3 g(NTOT / 128, NCLUST / 64);
    costs_kernel<<<g, 256, 0, stream>>>(inp, means, r2p, c2, costs);
    rowpass_kernel<<<NTOT / 8, 256, 0, stream>>>(costs, intra, pred, pIntra, pAcc);
    variance_kernel<<<1, 256, 0, stream>>>(pIntra, pAcc, scal, outAcc);
    loss_kernel<<<NTOT / 8, 256, 0, stream>>>(costs, target, intra, pred, scal,
                                              outLoss, outPred, pLoss);
    finalize_kernel<<<1, 256, 0, stream>>>(pLoss, out);
}